// GNNs_32049045962863
// MI455X (gfx1250) — compile-verified
//
#include <hip/hip_runtime.h>

// Problem constants (from reference)
constexpr int CH = 128;   // channels
constexpr int NG = 64;    // graphs
constexpr int NL = 3;     // layers
#define NEG_SLOPE 0.01f
#define GN_EPS 1e-5f

typedef __attribute__((ext_vector_type(2))) float v2f;
typedef __attribute__((ext_vector_type(8))) float v8f;

__device__ __forceinline__ float4 f4add(float4 a, float4 b) {
  return make_float4(a.x + b.x, a.y + b.y, a.z + b.z, a.w + b.w);
}

// ----------------- setup kernels -----------------
__global__ void k_init(float* deg, int* cnt, int N) {
  int i = blockIdx.x * blockDim.x + threadIdx.x;
  if (i < N) deg[i] = 1.0f;          // +1 self loop
  if (i < NG) cnt[i] = 0;
}

__global__ void k_edge_deg(const int* __restrict__ dst, float* deg, int E) {
  int i = blockIdx.x * blockDim.x + threadIdx.x;
  if (i < E) atomicAdd(&deg[dst[i]], 1.0f);
}

__global__ void k_node_cnt(const int* __restrict__ batch, int* cnt, int N) {
  int i = blockIdx.x * blockDim.x + threadIdx.x;
  if (i < N) atomicAdd(&cnt[batch[i]], 1);
}

__global__ void k_dinv(float* deg, int N) {   // deg -> rsqrt(deg) in place
  int i = blockIdx.x * blockDim.x + threadIdx.x;
  if (i < N) deg[i] = rsqrtf(deg[i]);
}

__global__ void k_scan(const int* __restrict__ cnt, int* starts, float* cinv) {
  if (blockIdx.x == 0 && threadIdx.x == 0) {
    int acc = 0;
    for (int g = 0; g < NG; ++g) {
      starts[g] = acc;
      acc += cnt[g];
      cinv[g] = 1.0f / (float)(cnt[g] > 0 ? cnt[g] : 1);
    }
    starts[NG] = acc;
  }
}

// ----------------- fp32 WMMA GEMM: XW = X @ W  (N x 128) @ (128 x 128) ---------
// One wave per 16-row tile. W staged in LDS (64KB), shared by 8 waves/block.
// V_WMMA_F32_16X16X4_F32 layouts (ISA 7.12.2):
//   A 16x4: lane%16 = M, lane/16 selects K-pair {2kh, 2kh+1} across 2 VGPRs
//   B 4x16: lane%16 = N, lane/16 selects K-pair (mirrors A)
//   C/D 16x16: VGPR r -> M = r + 8*(lane/16), N = lane%16
__global__ __launch_bounds__(256) void k_gemm(const float* __restrict__ X,
                                              const float* __restrict__ Wm,
                                              float* __restrict__ XW,
                                              int ntiles) {
  __shared__ float lW[CH * CH];
  {
    float4* d4 = (float4*)lW;
    const float4* s4 = (const float4*)Wm;
    for (int i = threadIdx.x; i < CH * CH / 4; i += 256) d4[i] = s4[i];
  }
  __syncthreads();

  const int lane = threadIdx.x & 31;
  const int tile = blockIdx.x * 8 + (threadIdx.x >> 5);
  if (tile >= ntiles) return;   // whole wave uniform -> EXEC all-1s for WMMA

  const int kh  = lane >> 4;        // K half select
  const int col = lane & 15;        // M for A loads / N for B+C
  const int mrow = tile * 16 + col;

  // Preload this wave's A fragments for all 32 K-steps (64 VGPRs)
  v2f a[32];
  const float* xr = X + (size_t)mrow * CH;
#pragma unroll
  for (int s = 0; s < 32; ++s)
    a[s] = *(const v2f*)(xr + 4 * s + 2 * kh);

  for (int n = 0; n < 8; ++n) {           // 8 column tiles of 16
    v8f acc = {0.f, 0.f, 0.f, 0.f, 0.f, 0.f, 0.f, 0.f};
    const float* wb = lW + n * 16 + col;
#pragma unroll
    for (int s = 0; s < 32; ++s) {
      const int k0 = 4 * s + 2 * kh;
      v2f b;
      b.x = wb[(size_t)k0 * CH];
      b.y = wb[(size_t)(k0 + 1) * CH];
      acc = __builtin_amdgcn_wmma_f32_16x16x4_f32(
          false, a[s], false, b, (short)0, acc, false, false);
    }
    float* o = XW + (size_t)(tile * 16 + 8 * kh) * CH + n * 16 + col;
#pragma unroll
    for (int r = 0; r < 8; ++r) o[(size_t)r * CH] = acc[r];
  }
}

// ----------------- self-loop + bias init of agg -----------------
__global__ void k_selfbias(const float* __restrict__ XW, const float* __restrict__ dinv,
                           const float* __restrict__ bias, float* __restrict__ AGG,
                           int N) {
  int i = blockIdx.x * blockDim.x + threadIdx.x;
  if (i >= N * 32) return;
  int n = i >> 5, cg = i & 31;
  float di = dinv[n];
  float sc = di * di;
  float4 v  = ((const float4*)XW)[(size_t)n * 32 + cg];
  float4 bb = ((const float4*)bias)[cg];
  ((float4*)AGG)[(size_t)n * 32 + cg] =
      make_float4(v.x * sc + bb.x, v.y * sc + bb.y, v.z * sc + bb.z, v.w * sc + bb.w);
}

// ----------------- edge scatter: agg[dst] += xw[src] * dinv[src]*dinv[dst] ------
__global__ void k_edge(const float* __restrict__ XW, const float* __restrict__ dinv,
                       const int* __restrict__ src, const int* __restrict__ dst,
                       float* __restrict__ AGG, int E) {
  int e = blockIdx.x * 8 + (threadIdx.x >> 5);
  if (e >= E) return;
  int lane = threadIdx.x & 31;
  int s = src[e], d = dst[e];
  float norm = dinv[s] * dinv[d];
  float4 v = ((const float4*)XW)[(size_t)s * 32 + lane];
  float* ad = AGG + (size_t)d * CH + lane * 4;
  atomicAdd(ad + 0, v.x * norm);
  atomicAdd(ad + 1, v.y * norm);
  atomicAdd(ad + 2, v.z * norm);
  atomicAdd(ad + 3, v.w * norm);
}

// ----------------- GraphNorm + LeakyReLU, in place over agg --------------------
// One block per graph (batch is sorted -> contiguous node ranges). No atomics.
__global__ __launch_bounds__(256) void k_gnorm(float* __restrict__ agg,
                                               const int* __restrict__ starts,
                                               const float* __restrict__ cinv,
                                               const float* __restrict__ gw,
                                               const float* __restrict__ gb,
                                               const float* __restrict__ gms,
                                               float* __restrict__ xfin) {
  __shared__ float4 red[256];
  __shared__ float4 smean[32];
  __shared__ float4 sistd[32];

  const int g  = blockIdx.x;
  const int t  = threadIdx.x;
  const int cg = t & 31;    // channel group (4 channels each)
  const int ns = t >> 5;    // node stride slot (8)
  const int s0 = starts[g], s1 = starts[g + 1];
  const float ci = cinv[g];
  float4* A = (float4*)agg;

  // pass 1: per-graph mean
  float4 acc = make_float4(0.f, 0.f, 0.f, 0.f);
  for (int n = s0 + ns; n < s1; n += 8)
    acc = f4add(acc, A[(size_t)n * 32 + cg]);
  red[t] = acc;
  __syncthreads();
  if (t < 128) red[t] = f4add(red[t], red[t + 128]);
  __syncthreads();
  if (t < 64) red[t] = f4add(red[t], red[t + 64]);
  __syncthreads();
  if (t < 32) {
    float4 m = f4add(red[t], red[t + 32]);
    smean[t] = make_float4(m.x * ci, m.y * ci, m.z * ci, m.w * ci);
  }
  __syncthreads();

  const float4 mean = smean[cg];
  const float4 ms   = ((const float4*)gms)[cg];
  const float4 sub  = make_float4(mean.x * ms.x, mean.y * ms.y,
                                  mean.z * ms.z, mean.w * ms.w);

  // pass 2: variance of (x - mean*mean_scale)
  acc = make_float4(0.f, 0.f, 0.f, 0.f);
  for (int n = s0 + ns; n < s1; n += 8) {
    float4 v = A[(size_t)n * 32 + cg];
    float4 o = make_float4(v.x - sub.x, v.y - sub.y, v.z - sub.z, v.w - sub.w);
    acc = f4add(acc, make_float4(o.x * o.x, o.y * o.y, o.z * o.z, o.w * o.w));
  }
  red[t] = acc;
  __syncthreads();
  if (t < 128) red[t] = f4add(red[t], red[t + 128]);
  __syncthreads();
  if (t < 64) red[t] = f4add(red[t], red[t + 64]);
  __syncthreads();
  if (t < 32) {
    float4 vv = f4add(red[t], red[t + 32]);
    sistd[t] = make_float4(rsqrtf(vv.x * ci + GN_EPS), rsqrtf(vv.y * ci + GN_EPS),
                           rsqrtf(vv.z * ci + GN_EPS), rsqrtf(vv.w * ci + GN_EPS));
  }
  __syncthreads();

  const float4 istd = sistd[cg];
  const float4 w4 = ((const float4*)gw)[cg];
  const float4 b4 = ((const float4*)gb)[cg];

  // pass 3: normalize + affine + leaky relu, write in place (+ final output copy)
  for (int n = s0 + ns; n < s1; n += 8) {
    float4 v = A[(size_t)n * 32 + cg];
    float4 o;
    o.x = (v.x - sub.x) * istd.x * w4.x + b4.x;
    o.y = (v.y - sub.y) * istd.y * w4.y + b4.y;
    o.z = (v.z - sub.z) * istd.z * w4.z + b4.z;
    o.w = (v.w - sub.w) * istd.w * w4.w + b4.w;
    o.x = o.x > 0.f ? o.x : NEG_SLOPE * o.x;
    o.y = o.y > 0.f ? o.y : NEG_SLOPE * o.y;
    o.z = o.z > 0.f ? o.z : NEG_SLOPE * o.z;
    o.w = o.w > 0.f ? o.w : NEG_SLOPE * o.w;
    A[(size_t)n * 32 + cg] = o;
    if (xfin) ((float4*)xfin)[(size_t)n * 32 + cg] = o;
  }
}

// ----------------- launch -----------------
extern "C" void kernel_launch(void* const* d_in, const int* in_sizes, int n_in,
                              void* d_out, int out_size, void* d_ws, size_t ws_size,
                              hipStream_t stream) {
  const float* x0    = (const float*)d_in[0];
  const int*   ei    = (const int*)d_in[1];
  const int*   batch = (const int*)d_in[2];
  const float* Wp    = (const float*)d_in[3];
  const float* bp    = (const float*)d_in[4];
  const float* gwp   = (const float*)d_in[5];
  const float* gbp   = (const float*)d_in[6];
  const float* gmsp  = (const float*)d_in[7];

  const int N = in_sizes[0] / CH;
  const int E = in_sizes[1] / 2;
  const int* src = ei;
  const int* dst = ei + E;

  float* out  = (float*)d_out;
  float* xfin = out;                      // x_final [N,C]
  float* hist = out + (size_t)N * CH;     // history [L,N,C]

  // workspace: xw [N*C] | deg/dinv [N] | cinv [NG] | cnt [NG] | starts [NG+1]
  float* xw     = (float*)d_ws;
  float* deg    = xw + (size_t)N * CH;
  float* cinv   = deg + N;
  int*   cnt    = (int*)(cinv + NG);
  int*   starts = cnt + NG;

  const int TPB = 256;
  k_init<<<(N + TPB - 1) / TPB, TPB, 0, stream>>>(deg, cnt, N);
  k_edge_deg<<<(E + TPB - 1) / TPB, TPB, 0, stream>>>(dst, deg, E);
  k_node_cnt<<<(N + TPB - 1) / TPB, TPB, 0, stream>>>(batch, cnt, N);
  k_dinv<<<(N + TPB - 1) / TPB, TPB, 0, stream>>>(deg, N);  // deg -> dinv
  k_scan<<<1, 1, 0, stream>>>(cnt, starts, cinv);

  const int ntiles = N / 16;  // N = 100000 is a multiple of 16
  for (int l = 0; l < NL; ++l) {
    const float* xin = (l == 0) ? x0 : hist + (size_t)(l - 1) * N * CH;
    float* agg = hist + (size_t)l * N * CH;   // agg lives in the history slot
    k_gemm<<<(ntiles + 7) / 8, 256, 0, stream>>>(xin, Wp + (size_t)l * CH * CH, xw, ntiles);
    k_selfbias<<<(N * 32 + TPB - 1) / TPB, TPB, 0, stream>>>(xw, deg, bp + l * CH, agg, N);
    k_edge<<<(E + 7) / 8, 256, 0, stream>>>(xw, deg, src, dst, agg, E);
    k_gnorm<<<NG, 256, 0, stream>>>(agg, starts, cinv, gwp + l * CH, gbp + l * CH,
                                    gmsp + l * CH, (l == NL - 1) ? xfin : nullptr);
  }
}